// NeighborSearchLayer_58445914964180
// MI455X (gfx1250) — compile-verified
//
#include <hip/hip_runtime.h>
#include <hip/hip_bf16.h>

// Fixed-radius neighbor search (R = 0.1) on 16384 x 16384 points via
// V_WMMA_F32_16X16X4_F32. Per 16x16 tile (256 pairs), one WMMA computes
//   D_ij = |q_i|^2 - 2 q_i.p_j       (C = inline 0)
//   A row i = [-2x_i, -2y_i, -2z_i, |q_i|^2]   (16x4, K padded 3->4)
//   B col j = [ x_j,   y_j,   z_j,  1      ]   (4x16)
// neighbor iff D_ij <= thr_j where thr_j = R^2 - |p_j|^2 (per-lane VGPR).
// Passes: fill(-1) -> per-row counts (WMMA + per-lane accumulate + shfl_xor
// reduce) -> scan -> ordered emit (WMMA + ballot ranks, empty-tile skip).

#define NPTS      16384
#define NTILES    (NPTS / 16)     // 1024 column tiles / row blocks
#define MAX_EDGES 4194304
#define R2        0.01f
#define WAVES_PER_BLOCK 8         // 256 threads = 8 wave32

typedef float v2f __attribute__((ext_vector_type(2)));
typedef float v8f __attribute__((ext_vector_type(8)));

__device__ __forceinline__ unsigned ballot32(bool p) {
  return __builtin_amdgcn_ballot_w32(p);
}

// ---- A fragment: 16x4 f32. lanes 0-15: (K0,K1) of row M=lane; lanes 16-31: (K2,K3).
__device__ __forceinline__ v2f make_a_frag(const float* __restrict__ qry,
                                           int rbase, int lane, int lrow) {
  const float* q = qry + (size_t)(rbase + lrow) * 3;
  float qx = q[0], qy = q[1], qz = q[2];
  float qa2 = qx * qx + qy * qy + qz * qz;
  v2f a;
  a.x = (lane < 16) ? (-2.0f * qx) : (-2.0f * qz);
  a.y = (lane < 16) ? (-2.0f * qy) : qa2;
  return a;
}

// ---- B fragment (4x16) + per-lane threshold thr = R^2 - |p|^2 for tile jt.
__device__ __forceinline__ void make_b_thr(const float* __restrict__ inp,
                                           int jt, int lane, int lrow,
                                           v2f& b, float& thr) {
  const float* p = inp + (size_t)(jt * 16 + lrow) * 3;
  float px = p[0], py = p[1], pz = p[2];
  b.x = (lane < 16) ? px : pz;    // VGPR0: K=0 (lo half) / K=2 (hi half)
  b.y = (lane < 16) ? py : 1.0f;  // VGPR1: K=1 (lo half) / K=3 (hi half)
  thr = __builtin_fmaf(-px, px,
        __builtin_fmaf(-py, py,
        __builtin_fmaf(-pz, pz, R2)));
}

// ---------------- Pass 0: preset neighbors_index to -1 ----------------
__global__ __launch_bounds__(256) void fill_kernel(int* __restrict__ nbr) {
  int i = blockIdx.x * blockDim.x + threadIdx.x;  // one int4 per thread
  reinterpret_cast<int4*>(nbr)[i] = make_int4(-1, -1, -1, -1);
}

// ---------------- Pass 1: per-row neighbor counts ----------------
__global__ __launch_bounds__(256) void nbr_count_kernel(
    const float* __restrict__ inp, const float* __restrict__ qry,
    int* __restrict__ counts) {
  const int lane = threadIdx.x & 31;
  const int lrow = lane & 15;
  const int rowblock = blockIdx.x * WAVES_PER_BLOCK + (threadIdx.x >> 5);
  const int rbase = rowblock << 4;

  v2f a = make_a_frag(qry, rbase, lane, lrow);
  const v8f czero = {};  // loop-invariant; SRC2 lowers to inline 0

  // Per-lane hit counters: cnt[v] counts hits of (row v [lanes 0-15] or
  // row v+8 [lanes 16-31], column = this lane's lrow) across all tiles.
  unsigned cnt[8] = {0, 0, 0, 0, 0, 0, 0, 0};

  for (int jt = 0; jt < NTILES; ++jt) {
    v2f b; float thr;
    make_b_thr(inp, jt, lane, lrow, b, thr);
    v8f d = __builtin_amdgcn_wmma_f32_16x16x4_f32(
        false, a, false, b, (short)0, czero, false, false);
#pragma unroll
    for (int v = 0; v < 8; ++v) cnt[v] += (d[v] <= thr) ? 1u : 0u;
  }

  // Reduce across the 16 lanes of each half (xor masks 1,2,4,8 never cross
  // the lane-16 boundary): lane 0 ends with rows rbase+v, lane 16 with
  // rows rbase+8+v. Lowers to ds_bpermute lane-permutes.
#pragma unroll
  for (int v = 0; v < 8; ++v) {
    unsigned s = cnt[v];
    s += (unsigned)__shfl_xor((int)s, 1);
    s += (unsigned)__shfl_xor((int)s, 2);
    s += (unsigned)__shfl_xor((int)s, 4);
    s += (unsigned)__shfl_xor((int)s, 8);
    cnt[v] = s;
  }
  if (lane == 0) {
#pragma unroll
    for (int v = 0; v < 8; ++v) counts[rbase + v] = (int)cnt[v];
  }
  if (lane == 16) {
#pragma unroll
    for (int v = 0; v < 8; ++v) counts[rbase + 8 + v] = (int)cnt[v];
  }
}

// ---------------- Pass 2: exclusive scan -> row_splits[16385] ----------------
__global__ __launch_bounds__(1024) void scan_kernel(const int* __restrict__ counts,
                                                    int* __restrict__ splits) {
  __shared__ int tsum[1024];
  const int t = threadIdx.x;
  int local[16];
  int s = 0;
#pragma unroll
  for (int i = 0; i < 16; ++i) { local[i] = counts[t * 16 + i]; s += local[i]; }
  tsum[t] = s;
  __syncthreads();
  for (int off = 1; off < 1024; off <<= 1) {
    int v = (t >= off) ? tsum[t - off] : 0;
    __syncthreads();
    if (t >= off) tsum[t] += v;
    __syncthreads();
  }
  int run = tsum[t] - s;  // exclusive base for this thread's 16 rows
  if (t == 0) splits[0] = 0;
#pragma unroll
  for (int i = 0; i < 16; ++i) { run += local[i]; splits[t * 16 + i + 1] = run; }
}

// ---------------- Pass 3: ordered neighbor emit ----------------
__global__ __launch_bounds__(256) void nbr_emit_kernel(
    const float* __restrict__ inp, const float* __restrict__ qry,
    const int* __restrict__ splits, int* __restrict__ nbr) {
  const int lane = threadIdx.x & 31;
  const int lrow = lane & 15;
  const int rowblock = blockIdx.x * WAVES_PER_BLOCK + (threadIdx.x >> 5);
  const int rbase = rowblock << 4;

  v2f a = make_a_frag(qry, rbase, lane, lrow);
  const v8f czero = {};

  // Running write offsets per row (wave-uniform since ballots are uniform).
  unsigned base_lo[8], base_hi[8];
#pragma unroll
  for (int v = 0; v < 8; ++v) {
    base_lo[v] = (unsigned)splits[rbase + v];
    base_hi[v] = (unsigned)splits[rbase + 8 + v];
  }

  for (int jt = 0; jt < NTILES; ++jt) {
    v2f b; float thr;
    make_b_thr(inp, jt, lane, lrow, b, thr);
    v8f d = __builtin_amdgcn_wmma_f32_16x16x4_f32(
        false, a, false, b, (short)0, czero, false, false);

    unsigned m[8];
#pragma unroll
    for (int v = 0; v < 8; ++v) m[v] = ballot32(d[v] <= thr);

    // ~0.4% pair density: most tiles are empty; skip all bookkeeping then.
    unsigned any = m[0] | m[1] | m[2] | m[3] | m[4] | m[5] | m[6] | m[7];
    if (any) {
      const int col = jt * 16 + lrow;
      const unsigned below = (1u << lrow) - 1u;
#pragma unroll
      for (int v = 0; v < 8; ++v) {
        unsigned lo = m[v] & 0xFFFFu;         // row rbase+v, cols jt*16..+15
        unsigned hi = m[v] >> 16;             // row rbase+8+v
        bool act = (m[v] >> lane) & 1u;
        unsigned pos = (lane < 16)
                           ? (base_lo[v] + (unsigned)__popc(lo & below))
                           : (base_hi[v] + (unsigned)__popc(hi & below));
        if (act && pos < MAX_EDGES) nbr[pos] = col;  // ascending (row,col)
        base_lo[v] += (unsigned)__popc(lo);
        base_hi[v] += (unsigned)__popc(hi);
      }
    }
  }
}

extern "C" void kernel_launch(void* const* d_in, const int* in_sizes, int n_in,
                              void* d_out, int out_size, void* d_ws, size_t ws_size,
                              hipStream_t stream) {
  const float* inp = (const float*)d_in[0];  // inp_positions [16384,3] (neighbors)
  const float* qry = (const float*)d_in[1];  // out_positions [16384,3] (queries)
  int* out    = (int*)d_out;
  int* nbr    = out;              // neighbors_index [MAX_EDGES] (int32 bits)
  int* splits = out + MAX_EDGES;  // row_splits [16385]
  int* counts = (int*)d_ws;       // 16384 ints of scratch

  fill_kernel<<<MAX_EDGES / (256 * 4), 256, 0, stream>>>(nbr);
  nbr_count_kernel<<<NTILES / WAVES_PER_BLOCK, 256, 0, stream>>>(inp, qry, counts);
  scan_kernel<<<1, 1024, 0, stream>>>(counts, splits);
  nbr_emit_kernel<<<NTILES / WAVES_PER_BLOCK, 256, 0, stream>>>(inp, qry, splits, nbr);
}